// LocalFeatureCluster_90159953478388
// MI455X (gfx1250) — compile-verified
//
#include <hip/hip_runtime.h>
#include <hip/hip_fp16.h>

typedef float v2f __attribute__((ext_vector_type(2)));
typedef float v8f __attribute__((ext_vector_type(8)));

static constexpr int B_ = 8, C_ = 512, H_ = 48, W_ = 48;
static constexpr int N_ = H_ * W_;      // 2304
static constexpr int K_ = 691;          // max(2, int(N*0.3))
static constexpr int KP_ = 704;         // K padded to multiple of 16
static constexpr int KM_ITERS = 10;
static constexpr int LDS_STRIDE = 516;  // 512 + 4: A-fetch hits 32 disjoint bank pairs

__device__ __forceinline__ float gelu_exact(float x) {
  return 0.5f * x * (1.0f + erff(x * 0.7071067811865475f));
}
__device__ __forceinline__ float sigmoidf_(float x) { return 1.0f / (1.0f + expf(-x)); }

// ---------------------------------------------------------------- utilities
__global__ void k_zero(float* p, int n) {
  int i = blockIdx.x * 256 + threadIdx.x;
  if (i < n) p[i] = 0.f;
}

// ------------------------------------------------ position encoder, stage 1
__global__ void k_prep16(const float* __restrict__ te_w1, const float* __restrict__ te_b1,
                         const float* __restrict__ fe_w1, const float* __restrict__ fe_b1,
                         float* __restrict__ t1g, float* __restrict__ f1g) {
  int i = blockIdx.x * 256 + threadIdx.x;
  if (i >= 16 * N_) return;
  int ch = i / N_, pix = i % N_;
  int y = pix / W_, x = pix % W_;
  float xs = -1.f + 2.f * (float)x / (float)(W_ - 1);
  float ys = -1.f + 2.f * (float)y / (float)(H_ - 1);
  float tw = 0.f, fw = 0.f;
  #pragma unroll
  for (int t = 0; t < 3; ++t) {
    if ((unsigned)(y + t - 1) < (unsigned)H_) tw += te_w1[ch * 3 + t];
    if ((unsigned)(x + t - 1) < (unsigned)W_) fw += fe_w1[ch * 3 + t];
  }
  t1g[i] = gelu_exact(te_b1[ch] + xs * tw);
  f1g[i] = gelu_exact(fe_b1[ch] + ys * fw);
}

// ------------------------------------------------ position encoder, stage 2
__global__ void k_tf(const float* __restrict__ te_w2, const float* __restrict__ te_b2,
                     const float* __restrict__ fe_w2, const float* __restrict__ fe_b2,
                     const float* __restrict__ t1g, const float* __restrict__ f1g,
                     float* __restrict__ tb, float* __restrict__ fb) {
  int idx = blockIdx.x * 256 + threadIdx.x;
  if (idx >= C_ * N_) return;
  int c = idx / N_, pix = idx % N_;
  int y = pix / W_, x = pix % W_;
  float ta = te_b2[c], fa = fe_b2[c];
  for (int i = 0; i < 16; ++i) {
    #pragma unroll
    for (int t = 0; t < 3; ++t) {
      int yy = y + t - 1;
      if ((unsigned)yy < (unsigned)H_)
        ta += te_w2[(c * 16 + i) * 3 + t] * t1g[i * N_ + yy * W_ + x];
      int xx = x + t - 1;
      if ((unsigned)xx < (unsigned)W_)
        fa += fe_w2[(c * 16 + i) * 3 + t] * f1g[i * N_ + y * W_ + xx];
    }
  }
  tb[idx] = ta;
  fb[idx] = fa;
}

// ------------------------------------------------ gate (per-pixel 1x1 MLP)
__global__ __launch_bounds__(256) void k_gate(const float* __restrict__ g_w1, const float* __restrict__ g_b1,
                                              const float* __restrict__ g_w2, const float* __restrict__ g_b2,
                                              const float* __restrict__ tb, const float* __restrict__ fb,
                                              float* __restrict__ pos) {
  __shared__ float comb[1024];
  __shared__ float hid[256];
  int pix = blockIdx.x, t = threadIdx.x;
  for (int j = t; j < 512; j += 256) {
    comb[j] = tb[j * N_ + pix];
    comb[512 + j] = fb[j * N_ + pix];
  }
  __syncthreads();
  float acc = g_b1[t];
  for (int i = 0; i < 1024; ++i) acc += comb[i] * g_w1[t * 1024 + i];
  hid[t] = fmaxf(acc, 0.f);
  __syncthreads();
  for (int cc = t; cc < 512; cc += 256) {
    float a2 = g_b2[cc];
    for (int i = 0; i < 256; ++i) a2 += hid[i] * g_w2[cc * 256 + i];
    float g = sigmoidf_(a2);
    pos[cc * N_ + pix] = g * comb[cc] + (1.f - g) * comb[512 + cc];
  }
}

// flat[b][n][c] = features[b][c][n] + pos[c][n]
__global__ void k_flat(const float* __restrict__ feat, const float* __restrict__ pos,
                       float* __restrict__ flat) {
  int i = blockIdx.x * 256 + threadIdx.x;
  if (i >= B_ * N_ * C_) return;
  int c = i % C_;
  int n = (i / C_) % N_;
  int b = i / (C_ * N_);
  flat[i] = feat[((size_t)b * C_ + c) * N_ + n] + pos[c * N_ + n];
}

// sim[p][q] = exp(-||pos_p - pos_q||)  stored f16 (batch-independent)
__global__ void k_sim(__half* __restrict__ simH) {
  int i = blockIdx.x * 256 + threadIdx.x;
  if (i >= N_ * N_) return;
  int p = i / N_, q = i % N_;
  float dy = (float)(p / W_ - q / W_);
  float dx = (float)(p % W_ - q % W_);
  simH[i] = __float2half(expf(-sqrtf(dy * dy + dx * dx)));
}

__global__ __launch_bounds__(256) void k_gmean(const float* __restrict__ feat, float* __restrict__ gf) {
  __shared__ float red[256];
  int bc = blockIdx.x, t = threadIdx.x;
  const float* src = feat + (size_t)bc * N_;
  float s = 0.f;
  for (int n = t; n < N_; n += 256) s += src[n];
  red[t] = s;
  __syncthreads();
  for (int o = 128; o > 0; o >>= 1) { if (t < o) red[t] += red[t + o]; __syncthreads(); }
  if (t == 0) gf[bc] = red[0] / (float)N_;
}

__global__ void k_initcent(const float* __restrict__ flat, float* __restrict__ cent) {
  int i = blockIdx.x * 256 + threadIdx.x;
  if (i >= B_ * KP_ * C_) return;
  int c = i % C_;
  int k = (i / C_) % KP_;
  int b = i / (C_ * KP_);
  float v = 0.f;
  if (k < K_) {
    int idx = (int)((double)k * (double)(N_ - 1) / (double)(K_ - 1));
    v = flat[((size_t)b * N_ + idx) * C_ + c];
  }
  cent[i] = v;
}

__global__ void k_cnorm(const float* __restrict__ cent, float* __restrict__ cn) {
  int i = blockIdx.x * 256 + threadIdx.x;
  if (i >= B_ * KP_) return;
  if ((i % KP_) >= K_) { cn[i] = 3.0e38f; return; }
  const float* c = cent + (size_t)i * C_;
  float s = 0.f;
  for (int j = 0; j < C_; ++j) s += c[j] * c[j];
  cn[i] = s;
}

// ------------------------------------------------------------- WMMA kernels
// Fused distance GEMM + argmin. 4 waves/block; the 16-row X strip is staged
// into LDS once via async global->LDS (ASYNCcnt), each wave covers 11 of the
// 44 cluster tiles with v_wmma_f32_16x16x4_f32, partial argmins combined in LDS.
__global__ __launch_bounds__(128) void k_argmin(const float* __restrict__ flat,
                                                const float* __restrict__ cent,
                                                const float* __restrict__ cn,
                                                int* __restrict__ lab) {
  __shared__ float As[16 * LDS_STRIDE];   // ~33 KB, bank-conflict-free A fetches
  __shared__ float redv[4 * 16];
  __shared__ int   redi[4 * 16];

  int blk = blockIdx.x;
  int b = blk / (N_ / 16);
  int row0 = (blk % (N_ / 16)) * 16;
  int tid = threadIdx.x;
  int wave = tid >> 5, lane = tid & 31;
  int m = lane & 15, hi = lane >> 4;

  // ---- async stage: 16 rows x 512 floats, b128 per op, 16 ops per thread
  {
    const char* gbase = (const char*)(flat + ((size_t)(b * N_) + row0) * C_);
    #pragma unroll
    for (int j = 0; j < 16; ++j) {
      int idx = tid + j * 128;              // 2048 x 16B chunks
      int r = idx >> 7, ch = idx & 127;     // row, 16B chunk within row
      unsigned lds = (unsigned)(size_t)&As[r * LDS_STRIDE + ch * 4];
      unsigned long long ga = (unsigned long long)(gbase + ((size_t)r * C_ + ch * 4) * 4);
      asm volatile("global_load_async_to_lds_b128 %0, %1, off"
                   :: "v"(lds), "v"(ga) : "memory");
    }
    asm volatile("s_wait_asynccnt 0x0" ::: "memory");
  }
  __syncthreads();

  const float* cb = cent + (size_t)b * KP_ * C_;
  const float* cnb = cn + b * KP_;
  const float2* ap = reinterpret_cast<const float2*>(&As[m * LDS_STRIDE + 2 * hi]);

  float minv[8];
  int mini[8];
  #pragma unroll
  for (int r = 0; r < 8; ++r) { minv[r] = 3.0e38f; mini[r] = 0; }

  for (int ct = wave; ct < KP_ / 16; ct += 4) {
    int col0 = ct * 16;
    const float2* bp = reinterpret_cast<const float2*>(cb + (size_t)(col0 + m) * C_ + 2 * hi);
    v8f acc;
    #pragma unroll
    for (int r = 0; r < 8; ++r) acc[r] = 0.f;
    #pragma unroll 4
    for (int k = 0; k < C_ / 4; ++k) {
      float2 av = ap[2 * k];    // ds_load_b64, conflict-free
      float2 bv = bp[2 * k];    // global_load_b64, L2-resident
      v2f a; a[0] = av.x; a[1] = av.y;
      v2f bb; bb[0] = bv.x; bb[1] = bv.y;
      acc = __builtin_amdgcn_wmma_f32_16x16x4_f32(false, a, false, bb, (short)0, acc,
                                                  false, false);
    }
    float cnv = cnb[col0 + m];
    int cidx = col0 + m;
    #pragma unroll
    for (int r = 0; r < 8; ++r) {
      float sc = cnv - 2.0f * acc[r];       // reg r holds row r + 8*hi, col = m
      if (sc < minv[r]) { minv[r] = sc; mini[r] = cidx; }
    }
  }
  // per-wave reduce (min, idx) across the 16 lanes of each half-wave
  #pragma unroll
  for (int r = 0; r < 8; ++r) {
    float v = minv[r];
    int id = mini[r];
    for (int msk = 8; msk >= 1; msk >>= 1) {
      float ov = __shfl_xor(v, msk, 32);
      int oi = __shfl_xor(id, msk, 32);
      if (ov < v || (ov == v && oi < id)) { v = ov; id = oi; }
    }
    minv[r] = v;
    mini[r] = id;
  }
  if (m == 0) {
    #pragma unroll
    for (int r = 0; r < 8; ++r) {
      redv[wave * 16 + hi * 8 + r] = minv[r];
      redi[wave * 16 + hi * 8 + r] = mini[r];
    }
  }
  __syncthreads();
  // cross-wave combine: threads 0..15 pick the global argmin for their row
  if (tid < 16) {
    float best = redv[tid];
    int besti = redi[tid];
    #pragma unroll
    for (int w = 1; w < 4; ++w) {
      float v = redv[w * 16 + tid];
      int id = redi[w * 16 + tid];
      if (v < best || (v == best && id < besti)) { best = v; besti = id; }
    }
    lab[b * N_ + row0 + tid] = besti;
  }
}

// spatial_net layer 1: hidden[b][n][o] = flat[b][n] @ sp_w1 + sp_b1  (WMMA f32)
__global__ __launch_bounds__(32) void k_spgemm(const float* __restrict__ flat,
                                               const float* __restrict__ w1,
                                               const float* __restrict__ b1,
                                               float* __restrict__ hidden) {
  int blk = blockIdx.x;
  int ctile = blk & 15;
  int rt = (blk >> 4) % (N_ / 16);
  int b = blk / (16 * (N_ / 16));
  int row0 = rt * 16, col0 = ctile * 16;
  int lane = threadIdx.x;
  int m = lane & 15, hi = lane >> 4;
  const float2* ap = reinterpret_cast<const float2*>(flat + ((size_t)(b * N_) + row0 + m) * C_ + 2 * hi);
  v8f acc;
  #pragma unroll
  for (int r = 0; r < 8; ++r) acc[r] = 0.f;
  int col = col0 + m;
  for (int k = 0; k < C_ / 4; ++k) {
    float2 av = ap[2 * k];
    int kk = 4 * k + 2 * hi;
    v2f a; a[0] = av.x; a[1] = av.y;
    v2f bb; bb[0] = w1[kk * 256 + col]; bb[1] = w1[(kk + 1) * 256 + col];
    acc = __builtin_amdgcn_wmma_f32_16x16x4_f32(false, a, false, bb, (short)0, acc,
                                                false, false);
  }
  float bias = b1[col];
  float* hb = hidden + ((size_t)(b * N_) + row0) * 256 + col;
  #pragma unroll
  for (int r = 0; r < 8; ++r) hb[(size_t)(r + 8 * hi) * 256] = acc[r] + bias;
}

// ------------------------------------------------------------ k-means step
__global__ __launch_bounds__(64) void k_accum(const float* __restrict__ flat,
                                              const int* __restrict__ lab,
                                              float* __restrict__ acc, float* __restrict__ cnt) {
  int bn = blockIdx.x, t = threadIdx.x;
  int b = bn / N_;
  int l = lab[bn];
  const float* x = flat + (size_t)bn * C_;
  float* a = acc + ((size_t)b * KP_ + l) * C_;
  for (int c = t; c < C_; c += 64) atomicAdd(&a[c], x[c]);
  if (t == 0) atomicAdd(&cnt[b * KP_ + l], 1.0f);
}

__global__ void k_fincent(const float* __restrict__ acc, const float* __restrict__ cnt,
                          float* __restrict__ cent) {
  int i = blockIdx.x * 256 + threadIdx.x;
  if (i >= B_ * KP_ * C_) return;
  float c0 = cnt[i / C_];
  if (c0 > 0.f) cent[i] = acc[i] / fmaxf(c0, 1.f);
}

__global__ void k_count(const int* __restrict__ lab, float* __restrict__ cnt) {
  int i = blockIdx.x * 256 + threadIdx.x;
  if (i >= B_ * N_) return;
  atomicAdd(&cnt[(i / N_) * KP_ + lab[i]], 1.0f);
}

// ------------------------------------------ refine (sim@oh masked => row-sum)
__global__ __launch_bounds__(32) void k_srow(const __half* __restrict__ simH,
                                             const int* __restrict__ lab,
                                             const float* __restrict__ cnt,
                                             float* __restrict__ wq) {
  int bn = blockIdx.x;
  int b = bn / N_, n = bn % N_;
  int l = lab[bn];
  const int* labb = lab + b * N_;
  const __half* row = simH + (size_t)n * N_;
  float s = 0.f;
  for (int q = threadIdx.x; q < N_; q += 32)
    if (labb[q] == l) s += __half2float(row[q]);
  for (int o = 16; o > 0; o >>= 1) s += __shfl_down(s, o, 32);
  if (threadIdx.x == 0) wq[bn] = s / fmaxf(cnt[b * KP_ + l], 1.f);
}

__global__ __launch_bounds__(64) void k_waccum(const float* __restrict__ flat,
                                               const int* __restrict__ lab,
                                               const float* __restrict__ wq,
                                               float* __restrict__ acc, float* __restrict__ wsum) {
  int bn = blockIdx.x, t = threadIdx.x;
  int b = bn / N_;
  int l = lab[bn];
  float w = wq[bn];
  const float* x = flat + (size_t)bn * C_;
  float* a = acc + ((size_t)b * KP_ + l) * C_;
  for (int c = t; c < C_; c += 64) atomicAdd(&a[c], w * x[c]);
  if (t == 0) atomicAdd(&wsum[b * KP_ + l], w);
}

__global__ void k_refine(const float* __restrict__ acc, const float* __restrict__ wsum,
                         const float* __restrict__ cnt, const float* __restrict__ cent,
                         float* __restrict__ refc) {
  int i = blockIdx.x * 256 + threadIdx.x;
  if (i >= B_ * KP_ * C_) return;
  int bk = i / C_;
  refc[i] = (cnt[bk] > 0.f) ? acc[i] / (wsum[bk] + 1e-6f) : cent[i];
}

// ----------------------------------------------------------- tail of network
__global__ void k_sw(const float* __restrict__ hidden, const float* __restrict__ sp_w2,
                     const float* __restrict__ sp_b2, float* __restrict__ sw) {
  int i = blockIdx.x * 256 + threadIdx.x;
  if (i >= B_ * N_) return;
  const float* h = hidden + (size_t)i * 256;
  float s = sp_b2[0];
  for (int o = 0; o < 256; ++o) s += fmaxf(h[o], 0.f) * sp_w2[o];
  sw[i] = sigmoidf_(s);
}

__global__ __launch_bounds__(256) void k_lmean(const float* __restrict__ flat,
                                               const float* __restrict__ refc,
                                               const int* __restrict__ lab,
                                               const float* __restrict__ sw,
                                               float* __restrict__ lf) {
  __shared__ float red[256];
  int bc = blockIdx.x, t = threadIdx.x;
  int b = bc / C_, c = bc % C_;
  float s = 0.f;
  for (int n = t; n < N_; n += 256) {
    int l = lab[b * N_ + n];
    float w = sw[b * N_ + n];
    float cl = refc[((size_t)b * KP_ + l) * C_ + c];
    float fl = flat[((size_t)(b * N_ + n)) * C_ + c];
    s += w * cl + (1.f - w) * fl;
  }
  red[t] = s;
  __syncthreads();
  for (int o = 128; o > 0; o >>= 1) { if (t < o) red[t] += red[t + o]; __syncthreads(); }
  if (t == 0) lf[bc] = red[0] / (float)N_;
}

__global__ __launch_bounds__(256) void k_fusion(const float* __restrict__ gf, const float* __restrict__ lf,
                                                const float* __restrict__ fu_w1, const float* __restrict__ fu_b1,
                                                const float* __restrict__ fu_w2, const float* __restrict__ fu_b2,
                                                float* __restrict__ out) {
  __shared__ float comb[1024];
  __shared__ float hid[256];
  __shared__ float aval;
  int b = blockIdx.x, t = threadIdx.x;
  for (int j = t; j < 512; j += 256) {
    comb[j] = gf[b * C_ + j];
    comb[512 + j] = lf[b * C_ + j];
  }
  __syncthreads();
  float a2 = fu_b1[t];
  for (int i = 0; i < 1024; ++i) a2 += comb[i] * fu_w1[i * 256 + t];
  hid[t] = fmaxf(a2, 0.f) * fu_w2[t];
  __syncthreads();
  for (int o = 128; o > 0; o >>= 1) { if (t < o) hid[t] += hid[t + o]; __syncthreads(); }
  if (t == 0) aval = sigmoidf_(hid[0] + fu_b2[0]);
  __syncthreads();
  for (int cc = t; cc < 512; cc += 256)
    out[b * C_ + cc] = aval * gf[b * C_ + cc] + (1.f - aval) * lf[b * C_ + cc];
}

__global__ void k_copycent(const float* __restrict__ refc, float* __restrict__ oc) {
  int i = blockIdx.x * 256 + threadIdx.x;
  if (i >= B_ * K_ * C_) return;
  int c = i % C_;
  int k = (i / C_) % K_;
  int b = i / (C_ * K_);
  oc[i] = refc[((size_t)b * KP_ + k) * C_ + c];
}

// --------------------------------------------------------------------- host
extern "C" void kernel_launch(void* const* d_in, const int* in_sizes, int n_in,
                              void* d_out, int out_size, void* d_ws, size_t ws_size,
                              hipStream_t stream) {
  (void)in_sizes; (void)n_in; (void)out_size; (void)ws_size;
  const float* features = (const float*)d_in[0];
  const float* te_w1 = (const float*)d_in[1];
  const float* te_b1 = (const float*)d_in[2];
  const float* te_w2 = (const float*)d_in[3];
  const float* te_b2 = (const float*)d_in[4];
  const float* fe_w1 = (const float*)d_in[5];
  const float* fe_b1 = (const float*)d_in[6];
  const float* fe_w2 = (const float*)d_in[7];
  const float* fe_b2 = (const float*)d_in[8];
  const float* g_w1 = (const float*)d_in[9];
  const float* g_b1 = (const float*)d_in[10];
  const float* g_w2 = (const float*)d_in[11];
  const float* g_b2 = (const float*)d_in[12];
  const float* sp_w1 = (const float*)d_in[13];
  const float* sp_b1 = (const float*)d_in[14];
  const float* sp_w2 = (const float*)d_in[15];
  const float* sp_b2 = (const float*)d_in[16];
  const float* fu_w1 = (const float*)d_in[17];
  const float* fu_b1 = (const float*)d_in[18];
  const float* fu_w2 = (const float*)d_in[19];
  const float* fu_b2 = (const float*)d_in[20];

  float* out = (float*)d_out;
  float* outCent = out + B_ * C_;

  float* ws = (float*)d_ws;
  size_t off = 0;
  auto alloc = [&](size_t n) { float* p = ws + off; off += n; return p; };
  float* t1g = alloc((size_t)16 * N_);
  float* f1g = alloc((size_t)16 * N_);
  float* tb = alloc((size_t)C_ * N_);
  float* fb = alloc((size_t)C_ * N_);
  float* pos = alloc((size_t)C_ * N_);
  float* flat = alloc((size_t)B_ * N_ * C_);
  float* cent = alloc((size_t)B_ * KP_ * C_);
  float* acc = alloc((size_t)B_ * KP_ * C_);
  float* refc = alloc((size_t)B_ * KP_ * C_);
  float* cn = alloc((size_t)B_ * KP_);
  float* cnt = alloc((size_t)B_ * KP_);
  float* wsum = alloc((size_t)B_ * KP_);
  int* lab = (int*)alloc((size_t)B_ * N_);
  float* wq = alloc((size_t)B_ * N_);
  float* hidden = alloc((size_t)B_ * N_ * 256);
  float* sw = alloc((size_t)B_ * N_);
  float* gf = alloc((size_t)B_ * C_);
  float* lf = alloc((size_t)B_ * C_);
  __half* simH = (__half*)alloc(((size_t)N_ * N_ + 1) / 2);

  auto cdiv = [](int a, int b) { return (a + b - 1) / b; };

  k_prep16<<<cdiv(16 * N_, 256), 256, 0, stream>>>(te_w1, te_b1, fe_w1, fe_b1, t1g, f1g);
  k_tf<<<cdiv(C_ * N_, 256), 256, 0, stream>>>(te_w2, te_b2, fe_w2, fe_b2, t1g, f1g, tb, fb);
  k_gate<<<N_, 256, 0, stream>>>(g_w1, g_b1, g_w2, g_b2, tb, fb, pos);
  k_flat<<<cdiv(B_ * N_ * C_, 256), 256, 0, stream>>>(features, pos, flat);
  k_sim<<<cdiv(N_ * N_, 256), 256, 0, stream>>>(simH);
  k_gmean<<<B_ * C_, 256, 0, stream>>>(features, gf);

  k_initcent<<<cdiv(B_ * KP_ * C_, 256), 256, 0, stream>>>(flat, cent);
  const int nBKC = B_ * KP_ * C_, nBK = B_ * KP_;
  for (int it = 0; it < KM_ITERS; ++it) {
    k_cnorm<<<cdiv(nBK, 256), 256, 0, stream>>>(cent, cn);
    k_argmin<<<B_ * (N_ / 16), 128, 0, stream>>>(flat, cent, cn, lab);
    k_zero<<<cdiv(nBKC, 256), 256, 0, stream>>>(acc, nBKC);
    k_zero<<<cdiv(nBK, 256), 256, 0, stream>>>(cnt, nBK);
    k_accum<<<B_ * N_, 64, 0, stream>>>(flat, lab, acc, cnt);
    k_fincent<<<cdiv(nBKC, 256), 256, 0, stream>>>(acc, cnt, cent);
  }
  k_cnorm<<<cdiv(nBK, 256), 256, 0, stream>>>(cent, cn);
  k_argmin<<<B_ * (N_ / 16), 128, 0, stream>>>(flat, cent, cn, lab);
  k_zero<<<cdiv(nBK, 256), 256, 0, stream>>>(cnt, nBK);
  k_count<<<cdiv(B_ * N_, 256), 256, 0, stream>>>(lab, cnt);

  k_srow<<<B_ * N_, 32, 0, stream>>>(simH, lab, cnt, wq);
  k_zero<<<cdiv(nBKC, 256), 256, 0, stream>>>(acc, nBKC);
  k_zero<<<cdiv(nBK, 256), 256, 0, stream>>>(wsum, nBK);
  k_waccum<<<B_ * N_, 64, 0, stream>>>(flat, lab, wq, acc, wsum);
  k_refine<<<cdiv(nBKC, 256), 256, 0, stream>>>(acc, wsum, cnt, cent, refc);

  k_spgemm<<<B_ * (N_ / 16) * 16, 32, 0, stream>>>(flat, sp_w1, sp_b1, hidden);
  k_sw<<<cdiv(B_ * N_, 256), 256, 0, stream>>>(hidden, sp_w2, sp_b2, sw);
  k_lmean<<<B_ * C_, 256, 0, stream>>>(flat, refc, lab, sw, lf);
  k_fusion<<<B_, 256, 0, stream>>>(gf, lf, fu_w1, fu_b1, fu_w2, fu_b2, out);
  k_copycent<<<cdiv(B_ * K_ * C_, 256), 256, 0, stream>>>(refc, outCent);
}